// VarDecoder_12979391169448
// MI455X (gfx1250) — compile-verified
//
#include <hip/hip_runtime.h>
#include <math.h>

typedef __attribute__((ext_vector_type(2))) float v2f;
typedef __attribute__((ext_vector_type(8))) float v8f;

#define NDIM   256
#define CATIN  1024
#define HDIM   512
#define VDIM   32000
#define TDIM   8

__device__ __forceinline__ v8f wmma_f32(v2f a, v2f b, v8f c) {
  // D = A(16x4 f32) * B(4x16 f32) + C(16x16 f32)
  return __builtin_amdgcn_wmma_f32_16x16x4_f32(
      /*neg_a=*/false, a, /*neg_b=*/false, b,
      /*c_mod=*/(short)0, c, /*reuse_a=*/false, /*reuse_b=*/false);
}

__device__ __forceinline__ float sigf(float x) {
  return 1.0f / (1.0f + __expf(-x));
}

// ---------------------------------------------------------------------------
// Zero-fill (h0, c0) -- ws is poisoned 0xAA by the harness.
// ---------------------------------------------------------------------------
__global__ void zero_f32(float* __restrict__ p, int n) {
  int i = blockIdx.x * blockDim.x + threadIdx.x;
  if (i < n) p[i] = 0.0f;
}

// ---------------------------------------------------------------------------
// Out[M,Nc] = A[M,K] @ B[Nc,K]^T + bias[Nc]      (one 16x16 tile per wave)
// Grid must supply exactly (M/16)*(Nc/16) waves.
// ---------------------------------------------------------------------------
__global__ void gemm_nt_bias(const float* __restrict__ A, const float* __restrict__ B,
                             const float* __restrict__ bias, float* __restrict__ Out,
                             int M, int Nc, int K) {
  int wave = (blockIdx.x * blockDim.x + threadIdx.x) >> 5;
  int lane = threadIdx.x & 31;
  int ntn  = Nc >> 4;
  int m0 = (wave / ntn) << 4;
  int n0 = (wave % ntn) << 4;

  int lm = lane & 15;            // A row / B row (=output col) within tile
  int lk = (lane >> 4) << 1;     // k sub-offset 0 or 2
  const float* arow = A + (size_t)(m0 + lm) * K + lk;
  const float* brow = B + (size_t)(n0 + lm) * K + lk;

  v8f acc = {};
  for (int k = 0; k < K; k += 4) {
    v2f a = *(const v2f*)(arow + k);
    v2f b = *(const v2f*)(brow + k);
    acc = wmma_f32(a, b, acc);
  }

  int ncol  = n0 + (lane & 15);
  int mbase = m0 + ((lane >> 4) << 3);
  float bv = bias[ncol];
#pragma unroll
  for (int r = 0; r < 8; ++r)
    Out[(size_t)(mbase + r) * Nc + ncol] = acc[r] + bv;
}

// ---------------------------------------------------------------------------
// One LSTM step. Each wave owns a 16x16 (m, j) patch and computes the i,f,g,o
// gate tiles for it (4 WMMA accumulators sharing one A fragment), then does
// the cell update in registers. Grid = (N/16)*(H/16) waves = 512 waves.
// ---------------------------------------------------------------------------
__global__ void lstm_step(const float* __restrict__ Hprev, float* __restrict__ Hout,
                          float* __restrict__ Cst, const float* __restrict__ Xproj,
                          const float* __restrict__ Whh, const float* __restrict__ bhh) {
  int wave = (blockIdx.x * blockDim.x + threadIdx.x) >> 5;
  int lane = threadIdx.x & 31;
  const int jt = HDIM >> 4;            // 32 column tiles
  int m0 = (wave / jt) << 4;
  int j0 = (wave % jt) << 4;

  int lm = lane & 15;
  int lk = (lane >> 4) << 1;
  const float* arow = Hprev + (size_t)(m0 + lm) * HDIM + lk;
  const float* b0 = Whh + (size_t)(0 * HDIM + j0 + lm) * HDIM + lk;
  const float* b1 = Whh + (size_t)(1 * HDIM + j0 + lm) * HDIM + lk;
  const float* b2 = Whh + (size_t)(2 * HDIM + j0 + lm) * HDIM + lk;
  const float* b3 = Whh + (size_t)(3 * HDIM + j0 + lm) * HDIM + lk;

  v8f ai = {}, af = {}, ag = {}, ao = {};
  for (int k = 0; k < HDIM; k += 4) {
    v2f a = *(const v2f*)(arow + k);
    ai = wmma_f32(a, *(const v2f*)(b0 + k), ai);
    af = wmma_f32(a, *(const v2f*)(b1 + k), af);
    ag = wmma_f32(a, *(const v2f*)(b2 + k), ag);
    ao = wmma_f32(a, *(const v2f*)(b3 + k), ao);
  }

  int j     = j0 + (lane & 15);
  int mbase = m0 + ((lane >> 4) << 3);
  float bi = bhh[j];
  float bf = bhh[j + HDIM];
  float bg = bhh[j + 2 * HDIM];
  float bo = bhh[j + 3 * HDIM];

#pragma unroll
  for (int r = 0; r < 8; ++r) {
    int m = mbase + r;
    const float* xp = Xproj + (size_t)m * (4 * HDIM) + j;
    float gi = sigf(ai[r] + xp[0]        + bi);
    float gf = sigf(af[r] + xp[HDIM]     + bf);
    float gg = tanhf(ag[r] + xp[2*HDIM]  + bg);
    float go = sigf(ao[r] + xp[3*HDIM]   + bo);
    size_t idx = (size_t)m * HDIM + j;
    float c = gf * Cst[idx] + gi * gg;
    Cst[idx]  = c;
    Hout[idx] = go * tanhf(c);
  }
}

// ---------------------------------------------------------------------------
// Big output GEMM: Hall[T*N, H] @ Wout[V, H]^T + bout  ->  out[N, T, V].
// 16x64 per wave (A fragment reused across 4 B tiles). Rows r = t*N + n are
// scattered into the transposed [N, T, V] layout on store.
// Grid = (T*N/16) * (V/64) = 128 * 500 = 64000 waves.
// ---------------------------------------------------------------------------
__global__ void out_gemm(const float* __restrict__ Hall, const float* __restrict__ Wout,
                         const float* __restrict__ bout, float* __restrict__ Out) {
  const int NB = VDIM >> 6;            // 500 blocks of 64 columns
  int wave = (blockIdx.x * blockDim.x + threadIdx.x) >> 5;
  int lane = threadIdx.x & 31;
  int m0 = (wave / NB) << 4;
  int n0 = (wave % NB) << 6;

  int lm = lane & 15;
  int lk = (lane >> 4) << 1;
  const float* arow = Hall + (size_t)(m0 + lm) * HDIM + lk;
  const float* br0  = Wout + (size_t)(n0 + lm) * HDIM + lk;
  const float* br1  = br0 + (size_t)16 * HDIM;
  const float* br2  = br0 + (size_t)32 * HDIM;
  const float* br3  = br0 + (size_t)48 * HDIM;

  v8f c0 = {}, c1 = {}, c2 = {}, c3 = {};
  for (int k = 0; k < HDIM; k += 4) {
    v2f a = *(const v2f*)(arow + k);
    c0 = wmma_f32(a, *(const v2f*)(br0 + k), c0);
    c1 = wmma_f32(a, *(const v2f*)(br1 + k), c1);
    c2 = wmma_f32(a, *(const v2f*)(br2 + k), c2);
    c3 = wmma_f32(a, *(const v2f*)(br3 + k), c3);
  }

  int mbase = m0 + ((lane >> 4) << 3);
  int nc    = n0 + (lane & 15);
  float bv0 = bout[nc], bv1 = bout[nc + 16], bv2 = bout[nc + 32], bv3 = bout[nc + 48];
#pragma unroll
  for (int r = 0; r < 8; ++r) {
    int row = mbase + r;               // row = t*N + n
    int t = row >> 8;                  // / 256
    int n = row & 255;
    float* o = Out + ((size_t)n * TDIM + t) * VDIM;
    o[nc]      = c0[r] + bv0;
    o[nc + 16] = c1[r] + bv1;
    o[nc + 32] = c2[r] + bv2;
    o[nc + 48] = c3[r] + bv3;
  }
}

// ---------------------------------------------------------------------------
// Host launcher
// ---------------------------------------------------------------------------
extern "C" void kernel_launch(void* const* d_in, const int* in_sizes, int n_in,
                              void* d_out, int out_size, void* d_ws, size_t ws_size,
                              hipStream_t stream) {
  const float* context = (const float*)d_in[0];   // [N, CAT_IN]
  const float* W_merge = (const float*)d_in[1];   // [H, CAT_IN]
  const float* b_merge = (const float*)d_in[2];   // [H]
  const float* W_ih    = (const float*)d_in[3];   // [4H, H]
  const float* W_hh    = (const float*)d_in[4];   // [4H, H]
  const float* b_ih    = (const float*)d_in[5];   // [4H]
  const float* b_hh    = (const float*)d_in[6];   // [4H]
  const float* W_out   = (const float*)d_in[7];   // [V, H]
  const float* b_out   = (const float*)d_in[8];   // [V]
  float* out = (float*)d_out;                     // [N, T, V]

  // Workspace layout (floats): ctx | xproj | c | Hbuf[(T+1) slots]
  float* ws    = (float*)d_ws;
  float* ctx   = ws;                                   // N*H
  float* xproj = ctx   + (size_t)NDIM * HDIM;          // N*4H
  float* cbuf  = xproj + (size_t)NDIM * 4 * HDIM;      // N*H
  float* Hbuf  = cbuf  + (size_t)NDIM * HDIM;          // (T+1)*N*H  (slot 0 = h0 = 0)

  // 1) zero c and h0 (contiguous: cbuf then Hbuf slot 0)
  {
    int n = 2 * NDIM * HDIM;
    zero_f32<<<(n + 255) / 256, 256, 0, stream>>>(cbuf, n);
  }

  // 2) merge: ctx = context @ W_merge^T + b_merge   [256,512], K=1024
  {
    int waves = (NDIM / 16) * (HDIM / 16);             // 512
    gemm_nt_bias<<<waves / 8, 256, 0, stream>>>(context, W_merge, b_merge, ctx,
                                                NDIM, HDIM, CATIN);
  }

  // 3) xproj = ctx @ W_ih^T + b_ih                   [256,2048], K=512
  {
    int waves = (NDIM / 16) * (4 * HDIM / 16);         // 2048
    gemm_nt_bias<<<waves / 8, 256, 0, stream>>>(ctx, W_ih, b_ih, xproj,
                                                NDIM, 4 * HDIM, HDIM);
  }

  // 4) LSTM recurrence: 8 sequential steps (stream order = the dependency)
  {
    int waves = (NDIM / 16) * (HDIM / 16);             // 512 -> 64 blocks
    for (int t = 0; t < TDIM; ++t) {
      const float* hprev = Hbuf + (size_t)t * NDIM * HDIM;
      float*       hout  = Hbuf + (size_t)(t + 1) * NDIM * HDIM;
      lstm_step<<<waves / 8, 256, 0, stream>>>(hprev, hout, cbuf, xproj, W_hh, b_hh);
    }
  }

  // 5) fused output projection over all T*N rows -> out[N, T, V]
  {
    const float* Hall = Hbuf + (size_t)NDIM * HDIM;    // slots 1..T, contiguous
    int waves = (TDIM * NDIM / 16) * (VDIM / 64);      // 128 * 500 = 64000
    out_gemm<<<waves / 8, 256, 0, stream>>>(Hall, W_out, b_out, out);
  }
}